// YoloV1Loss_82806969467113
// MI455X (gfx1250) — compile-verified
//
#include <hip/hip_runtime.h>
#include <stdint.h>

// ---------------------------------------------------------------------------
// YOLOv1 loss, MI455X (gfx1250).
// Memory-bound: 193 MB in / 4 B out -> ~8.3 us at 23.3 TB/s HBM.
// Strategy: CDNA5 async global->LDS B128 streaming (ASYNCcnt), LDS fan-out of
// 120-byte rows (bank-conflict-free for wave32), wave32 shuffle reduction,
// deterministic 2-kernel tree reduction (no float atomics).
// ---------------------------------------------------------------------------

#define LEN 30
#define TILE_ROWS 256
#define TILE_FLOATS (TILE_ROWS * LEN)   // 7680 floats per array per tile
#define TILE_CHUNKS (TILE_FLOATS / 4)   // 1920 x 16B chunks

// __builtin_amdgcn_global_load_async_to_lds_b128 signature (from hipcc
// diagnostics): (int4 addrspace(1)*, int4 addrspace(3)*, imm offset, imm cpol)
typedef int v4i __attribute__((ext_vector_type(4)));
typedef __attribute__((address_space(1))) v4i as1_v4i;
typedef __attribute__((address_space(3))) v4i as3_v4i;

#if defined(__AMDGCN__) && __has_builtin(__builtin_amdgcn_global_load_async_to_lds_b128)
#define USE_ASYNC_LDS 1
#else
#define USE_ASYNC_LDS 0
#endif

__global__ __launch_bounds__(TILE_ROWS) void yolo_tile_kernel(
    const float* __restrict__ pred, const float* __restrict__ targ,
    float* __restrict__ partial)
{
    __shared__ float sp[TILE_FLOATS];           // staged pred tile
    __shared__ float st[TILE_FLOATS];           // staged target tile
    __shared__ float wred[TILE_ROWS / 32];

    const int tid = threadIdx.x;
    const long long base = (long long)blockIdx.x * TILE_FLOATS;

#if USE_ASYNC_LDS
    // CDNA5 async DMA: global -> LDS, 16B per lane per issue, no VGPR roundtrip.
    for (int i = tid; i < TILE_CHUNKS; i += TILE_ROWS) {
        __builtin_amdgcn_global_load_async_to_lds_b128(
            (as1_v4i*)(uintptr_t)(pred + base + (long long)i * 4),
            (as3_v4i*)(uint32_t)(uintptr_t)(sp + i * 4),
            0, 0);
        __builtin_amdgcn_global_load_async_to_lds_b128(
            (as1_v4i*)(uintptr_t)(targ + base + (long long)i * 4),
            (as3_v4i*)(uint32_t)(uintptr_t)(st + i * 4),
            0, 0);
    }
    asm volatile("s_wait_asynccnt 0" ::: "memory");
#else
    for (int i = tid; i < TILE_CHUNKS; i += TILE_ROWS) {
        float4 a = *(const float4*)(pred + base + (long long)i * 4);
        float4 b = *(const float4*)(targ + base + (long long)i * 4);
        *(float4*)(sp + i * 4) = a;
        *(float4*)(st + i * 4) = b;
    }
#endif
    __syncthreads();

    // ---- per-row loss -----------------------------------------------------
    const float* p = sp + tid * LEN;   // pred row, original layout
    const float* q = st + tid * LEN;   // target row, original layout

    // mask: T[:,4] = target[...,0]
    const float m = (q[0] > 0.0f) ? 1.0f : 0.0f;

    // class loss: sum_{k=10..29} (P[k]-T[k])^2 ; T[10:] == target[10:]
    float cls = 0.0f;
#pragma unroll
    for (int k = 10; k < 30; ++k) {
        float d = p[k] - q[k];
        cls += d * d;
    }

    // target box (permuted): tb = [t2, t3, t4, t5, t0]
    const float tx = q[2], ty = q[3], tw = q[4], th = q[5];

    float bx[2], by[2], bw[2], bh[2], bc[2], iou[2];
#pragma unroll
    for (int b = 0; b < 2; ++b) {
        bx[b] = p[5 * b + 0];
        by[b] = p[5 * b + 1];
        bw[b] = p[5 * b + 2];
        bh[b] = p[5 * b + 3];
        bc[b] = p[5 * b + 4];

        const float ptlx = bx[b] / 7.0f - bw[b] * 0.5f;
        const float pbrx = bx[b] / 7.0f + bw[b] * 0.5f;
        const float ptly = by[b] / 7.0f - bh[b] * 0.5f;
        const float pbry = by[b] / 7.0f + bh[b] * 0.5f;
        const float ttlx = tx / 7.0f - tw * 0.5f;
        const float tbrx = tx / 7.0f + tw * 0.5f;
        const float ttly = ty / 7.0f - th * 0.5f;
        const float tbry = ty / 7.0f + th * 0.5f;

        float wx = fminf(pbrx, tbrx) - fmaxf(ptlx, ttlx);
        float wy = fminf(pbry, tbry) - fmaxf(ptly, ttly);
        wx = fmaxf(wx, 0.0f);
        wy = fmaxf(wy, 0.0f);
        const float inner  = wx * wy;
        const float area_p = (pbrx - ptlx) * (pbry - ptly);
        const float area_t = (tbrx - ttlx) * (tbry - ttly);
        iou[b] = inner / (area_p + area_t - inner);
    }

    // argmax (first-max tie break like jnp.argmax) + responsible box
    const int   idx  = (iou[1] > iou[0]) ? 1 : 0;
    const float miou = fmaxf(iou[0], iou[1]);
    const float rx = bx[idx], ry = by[idx], rw = bw[idx], rh = bh[idx], rc = bc[idx];

    const float dx = rx - tx, dy = ry - ty;
    const float dw = sqrtf(rw) - sqrtf(tw);
    const float dh = sqrtf(rh) - sqrtf(th);
    const float coord = dx * dx + dy * dy + dw * dw + dh * dh;

    // noobj combined: (1-m)*sum_c2 + m*(sum_c2 - rc^2) = sum_c2 - m*rc^2
    const float sum_c2 = bc[0] * bc[0] + bc[1] * bc[1];
    const float noobj  = sum_c2 - m * rc * rc;
    const float dconf  = rc - miou;

    // LAMBDA_NOOBJ = 0.5*7/7 = 0.5 ; LAMBDA_COORD = 5.0
    float v = m * cls + 0.5f * noobj + 5.0f * (m * coord) + m * dconf * dconf;

    // ---- wave32 + block reduction ----------------------------------------
#pragma unroll
    for (int off = 16; off > 0; off >>= 1)
        v += __shfl_down(v, off, 32);
    const int lane = tid & 31;
    const int wid  = tid >> 5;
    if (lane == 0) wred[wid] = v;
    __syncthreads();
    if (tid == 0) {
        float s = 0.0f;
#pragma unroll
        for (int w = 0; w < TILE_ROWS / 32; ++w) s += wred[w];
        partial[blockIdx.x] = s;
    }
}

__global__ __launch_bounds__(256) void yolo_reduce_kernel(
    const float* __restrict__ partial, int n, float scale,
    float* __restrict__ out)
{
    __shared__ float wred[8];
    float v = 0.0f;
    for (int i = threadIdx.x; i < n; i += 256) v += partial[i];
#pragma unroll
    for (int off = 16; off > 0; off >>= 1)
        v += __shfl_down(v, off, 32);
    const int lane = threadIdx.x & 31;
    const int wid  = threadIdx.x >> 5;
    if (lane == 0) wred[wid] = v;
    __syncthreads();
    if (threadIdx.x == 0) {
        float s = 0.0f;
#pragma unroll
        for (int w = 0; w < 8; ++w) s += wred[w];
        out[0] = s * scale;
    }
}

extern "C" void kernel_launch(void* const* d_in, const int* in_sizes, int n_in,
                              void* d_out, int out_size, void* d_ws, size_t ws_size,
                              hipStream_t stream)
{
    (void)n_in; (void)out_size; (void)ws_size;
    const float* pred = (const float*)d_in[0];
    const float* targ = (const float*)d_in[1];
    float* out     = (float*)d_out;
    float* partial = (float*)d_ws;

    const int rows   = in_sizes[0] / LEN;          // 802816
    const int blocks = rows / TILE_ROWS;           // 3136 (exact)
    const float scale = 49.0f / (float)rows;       // == 1/N == 1/16384 exactly

    yolo_tile_kernel<<<blocks, TILE_ROWS, 0, stream>>>(pred, targ, partial);
    yolo_reduce_kernel<<<1, 256, 0, stream>>>(partial, blocks, scale, out);
}